// RGBDSimilarity_52055003628048
// MI455X (gfx1250) — compile-verified
//
#include <hip/hip_runtime.h>
#include <cmath>

typedef __attribute__((ext_vector_type(2))) float v2f;
typedef __attribute__((ext_vector_type(8))) float v8f;

#define C_IN 512
#define C8O  64
#define HWX  4096
#define NBAT 8

// ---------------------------------------------------------------------------
// Kernel 1: fused Q/K projection GEMM on the f32 WMMA pipe.
// One wave computes a 16(pixel) x 64(out-ch) tile of BOTH Q and K.
// D = A(16x4 f32, x-tile) * B(4x16 f32, weight-tile) + C, k-loop over 512.
// ---------------------------------------------------------------------------
__global__ __launch_bounds__(256) void rgbd_proj_wmma(
    const float* __restrict__ x,
    const float* __restrict__ Wq, const float* __restrict__ bq,
    const float* __restrict__ Wk, const float* __restrict__ bk,
    float* __restrict__ Qout, float* __restrict__ Kout)
{
    const int lane = threadIdx.x & 31;
    const int wave = threadIdx.x >> 5;
    const int tile = blockIdx.x * 8 + wave;        // 0..2047
    const int b    = tile >> 8;                    // batch
    const int l0   = (tile & 255) << 4;            // pixel base, step 16

    const int hl = lane >> 4;                      // lane half (0/1)
    const int q  = lane & 15;                      // intra-half lane

    // A-matrix source: x[b][c][l0+q], lanes 0-15 coalesced per row
    const float* xb = x + (size_t)b * C_IN * HWX + (size_t)(l0 + q);

    // Bias-initialized accumulators. C/D layout: VGPR r -> M = r + 8*hl, N = q+16j.
    v8f accQ[4], accK[4];
#pragma unroll
    for (int j = 0; j < 4; ++j) {
        const float vq = bq[16 * j + q];
        const float vk = bk[16 * j + q];
#pragma unroll
        for (int r = 0; r < 8; ++r) { accQ[j][r] = vq; accK[j][r] = vk; }
    }

#pragma unroll 2
    for (int c0 = 0; c0 < C_IN; c0 += 4) {
        // A 16x4 f32: VGPR0 = K {0|2}, VGPR1 = K {1|3} split by lane half.
        v2f A;
        A.x = xb[(size_t)(c0 + 2 * hl)     * HWX];
        A.y = xb[(size_t)(c0 + 1 + 2 * hl) * HWX];

#pragma unroll
        for (int j = 0; j < 4; ++j) {
            // B 4x16 f32: VGPR0 = row K {0|2}, VGPR1 = row K {1|3}, N = lane&15.
            // Adjacent k in a Wq row -> one aligned 8B load.
            const v2f Bq = *(const v2f*)(Wq + (size_t)(16 * j + q) * C_IN + c0 + 2 * hl);
            const v2f Bk = *(const v2f*)(Wk + (size_t)(16 * j + q) * C_IN + c0 + 2 * hl);
            accQ[j] = __builtin_amdgcn_wmma_f32_16x16x4_f32(
                false, A, false, Bq, (short)0, accQ[j], false, false);
            accK[j] = __builtin_amdgcn_wmma_f32_16x16x4_f32(
                false, A, false, Bk, (short)0, accK[j], false, false);
        }
    }

    // Store both projections in [b][l][o] layout (o contiguous, 256B rows).
    // These are re-read immediately by kernel 2 -> default (cache-resident) stores.
    float* qb = Qout + ((size_t)b * HWX + l0) * C8O;
    float* kb = Kout + ((size_t)b * HWX + l0) * C8O;
#pragma unroll
    for (int j = 0; j < 4; ++j) {
#pragma unroll
        for (int r = 0; r < 8; ++r) {
            const int row = r + 8 * hl;            // M (pixel within tile)
            qb[(size_t)row * C8O + 16 * j + q] = accQ[j][r];
            kb[(size_t)row * C8O + 16 * j + q] = accK[j][r];
        }
    }
}

// ---------------------------------------------------------------------------
// Kernel 2: 7x7 neighborhood energies. One thread per (pixel, dy) row:
// 229376 threads = 7168 waves (~7 waves/SIMD for latency hiding).
// q-vector (64 f32) in VGPRs; zero-padding semantics (pad -> rgb 0, |d_q|).
// Final outputs are write-once -> nontemporal stores (keep Q/K hot in L2).
// ---------------------------------------------------------------------------
__global__ __launch_bounds__(256) void rgbd_energy(
    const float* __restrict__ Q, const float* __restrict__ Kf,
    const float* __restrict__ d,
    float* __restrict__ rgb, float* __restrict__ dep)
{
    const int tid  = blockIdx.x * 256 + threadIdx.x;   // 0 .. B*HW*7-1
    const int dy   = tid % 7 - 3;                      // -3..3
    const int gid  = tid / 7;                          // pixel id 0..32767
    const int b    = gid >> 12;
    const int l    = gid & (HWX - 1);
    const int y    = l >> 6;
    const int xx   = l & 63;

    // Resident q-vector: 16 x float4
    const float4* q4 = (const float4*)(Q + (size_t)gid * C8O);
    float4 qreg[16];
#pragma unroll
    for (int i = 0; i < 16; ++i) qreg[i] = q4[i];

    const float  dq = d[gid];
    const float* kb = Kf + (size_t)b * HWX * C8O;
    const float* db = d + (size_t)b * HWX;

    const int yy = y + dy;
    const bool vy = (yy >= 0) && (yy < 64);

    float* rgbo = rgb + (size_t)gid * 49 + (dy + 3) * 7;
    float* depo = dep + (size_t)gid * 49 + (dy + 3) * 7;

#pragma unroll
    for (int dx = -3; dx <= 3; ++dx) {
        const int xp = xx + dx;
        float rv, dv;
        if (vy && xp >= 0 && xp < 64) {
            const int lp = yy * 64 + xp;
            const float4* k4 = (const float4*)(kb + (size_t)lp * C8O);
            float acc = 0.0f;
#pragma unroll
            for (int i = 0; i < 16; ++i) {
                const float4 kv = k4[i];
                acc = fmaf(qreg[i].x, kv.x, acc);
                acc = fmaf(qreg[i].y, kv.y, acc);
                acc = fmaf(qreg[i].z, kv.z, acc);
                acc = fmaf(qreg[i].w, kv.w, acc);
            }
            rv = acc;
            dv = fabsf(dq - db[lp]);
        } else {
            rv = 0.0f;          // zero-padded key => dot = 0
            dv = fabsf(dq);     // |d_q - 0|
        }
        __builtin_nontemporal_store(rv, rgbo + (dx + 3));
        __builtin_nontemporal_store(dv, depo + (dx + 3));
    }
}

// ---------------------------------------------------------------------------
extern "C" void kernel_launch(void* const* d_in, const int* in_sizes, int n_in,
                              void* d_out, int out_size, void* d_ws, size_t ws_size,
                              hipStream_t stream)
{
    const float* x  = (const float*)d_in[0];
    const float* d  = (const float*)d_in[1];
    const float* Wq = (const float*)d_in[2];
    const float* bq = (const float*)d_in[3];
    const float* Wk = (const float*)d_in[4];
    const float* bk = (const float*)d_in[5];

    float* rgb = (float*)d_out;                                // B*HW*49
    float* dep = rgb + (size_t)NBAT * HWX * 49;                // B*HW*49

    float* Qws = (float*)d_ws;                                 // B*HW*64 f32
    float* Kws = Qws + (size_t)NBAT * HWX * C8O;               // B*HW*64 f32

    // 2048 wave-tiles, 8 waves per 256-thread block
    rgbd_proj_wmma<<<256, 256, 0, stream>>>(x, Wq, bq, Wk, bk, Qws, Kws);
    // B*HW*7 threads, one (pixel, dy) row each: 896 blocks x 256
    rgbd_energy<<<896, 256, 0, stream>>>(Qws, Kws, d, rgb, dep);
}